// HDSNN1D_36335423324549
// MI455X (gfx1250) — compile-verified
//
#include <hip/hip_runtime.h>
#include <hip/hip_bf16.h>
#include <stdint.h>

typedef __attribute__((ext_vector_type(2))) float v2f;
typedef __attribute__((ext_vector_type(8))) float v8f;

#define N_PRE 32
#define N_NEU 128
#define NDEL  15
#define KDIM  480              // 32*15 (full K fwd; one 32-o chunk bwd)
#define WSTR  482              // padded A-tile stride: even (b64) + conflict-free
#define COLS  128              // t-columns per block
#define WIN   (COLS + NDEL - 1)  // 142-column input window
#define TSTR  34               // transposed-window stride: even + conflict-free

// ---------------------------------------------------------------------------
// Forward: logit[o,t] = sum_{i,d} a[i,t-15+d] * W[o,i,d];  p = sigmoid(logit)
// GEMM with K ordered k = d*32 + i. A = W tile (16x480), B = im2col(a).
// fp32 WMMA 16x16x4; 8 waves/block, one 16-col N-subtile each.
// Hot loop: 2x ds_load_b64 + 1x v_wmma per K-step (all-immediate offsets).
// ---------------------------------------------------------------------------
__global__ __launch_bounds__(256) void fwd_conv_wmma(
    const float* __restrict__ a, const float* __restrict__ W,
    const float* __restrict__ bconv, float* __restrict__ p_out,
    float* __restrict__ logit_out, int T)
{
    __shared__ float sW[16 * WSTR];       // 16 o-rows x 480, k-major
    __shared__ float sA[WIN * TSTR];      // transposed a window: [x][i]

    const int t0  = blockIdx.x * COLS;
    const int o0  = blockIdx.y * 16;
    const int tid = threadIdx.x;

    // stage W tile, permuted to k = d*32 + i
    for (int idx = tid; idx < 16 * KDIM; idx += 256) {
        int m = idx / KDIM, k = idx - m * KDIM;
        int i = k & 31, d = k >> 5;
        sW[m * WSTR + k] = W[(o0 + m) * KDIM + i * NDEL + d];
    }
    // stage a window transposed: sA[x*TSTR + i] = a[i, t0-15+x]
    for (int idx = tid; idx < N_PRE * WIN; idx += 256) {
        int i = idx / WIN, x = idx - i * WIN;     // consecutive tid -> consecutive x
        int g = t0 - NDEL + x;
        sA[x * TSTR + i] = (g >= 0 && g < T) ? a[(size_t)i * T + g] : 0.0f;
    }
    __syncthreads();

    const int lane = tid & 31;
    const int w    = tid >> 5;
    const int half = lane >> 4;          // 0: K 0,1   1: K 2,3 of each step
    const int ln   = lane & 15;
    const int col  = w * 16 + ln;        // local column (B/C N index)

    const float* aBase = sW + ln * WSTR + (half << 1);   // + (d*32 + ii)
    const float* bBase = sA + col * TSTR + (half << 1);  // + (d*TSTR + ii)

    v8f c = {};
    #pragma unroll
    for (int d = 0; d < NDEL; ++d) {
        #pragma unroll
        for (int ii = 0; ii < 32; ii += 4) {
            const v2f av = *(const v2f*)(aBase + (d << 5) + ii);
            const v2f bv = *(const v2f*)(bBase + d * TSTR + ii);
            c = __builtin_amdgcn_wmma_f32_16x16x4_f32(false, av, false, bv,
                                                      (short)0, c, false, false);
        }
    }

    const int t = t0 + col;
    if (t < T) {
        #pragma unroll
        for (int r = 0; r < 8; ++r) {
            const int o = o0 + r + (half << 3);
            const float logit = c[r] + bconv[o];
            const float p = 1.0f / (1.0f + __expf(-logit));
            p_out[(size_t)o * T + t]     = p;
            logit_out[(size_t)o * T + t] = logit;   // stashed for top-k select
        }
    }
}

// ---------------------------------------------------------------------------
// Exact top-k via 2-level 16-bit radix select on logits (monotonic w/ sigmoid)
// ---------------------------------------------------------------------------
__device__ __forceinline__ unsigned monokey(float v) {
    unsigned u = __float_as_uint(v);
    return u ^ ((u >> 31) ? 0xFFFFFFFFu : 0x80000000u);  // order-preserving
}

__global__ void k_zero(unsigned* __restrict__ p, int n) {
    int i = blockIdx.x * blockDim.x + threadIdx.x;
    if (i < n) p[i] = 0u;
}

__global__ void k_hist_hi(const float* __restrict__ v, unsigned* __restrict__ hist,
                          size_t n) {
    size_t stride = (size_t)gridDim.x * blockDim.x;
    for (size_t i = (size_t)blockIdx.x * blockDim.x + threadIdx.x; i < n; i += stride)
        atomicAdd(&hist[monokey(v[i]) >> 16], 1u);
}

__global__ void k_scan1(const unsigned* __restrict__ hist, unsigned* __restrict__ ctrl,
                        const int* __restrict__ kptr) {
    __shared__ unsigned csum[256];
    const int tid = threadIdx.x;
    unsigned s = 0;
    for (int b = 0; b < 256; ++b) s += hist[tid * 256 + b];
    csum[tid] = s;
    __syncthreads();
    if (tid == 0) {
        const unsigned k = (unsigned)*kptr;
        unsigned cum = 0;
        int c = 255;
        for (; c > 0; --c) { if (cum + csum[c] >= k) break; cum += csum[c]; }
        int b = c * 256 + 255;
        for (; b > c * 256; --b) { if (cum + hist[b] >= k) break; cum += hist[b]; }
        ctrl[0] = (unsigned)b;   // winning high-16 bucket
        ctrl[1] = cum;           // strictly-above count
    }
}

__global__ void k_hist_lo(const float* __restrict__ v, unsigned* __restrict__ hist,
                          const unsigned* __restrict__ ctrl, size_t n) {
    const unsigned B = ctrl[0];
    size_t stride = (size_t)gridDim.x * blockDim.x;
    for (size_t i = (size_t)blockIdx.x * blockDim.x + threadIdx.x; i < n; i += stride) {
        unsigned key = monokey(v[i]);
        if ((key >> 16) == B) atomicAdd(&hist[key & 0xFFFFu], 1u);
    }
}

__global__ void k_scan2(const unsigned* __restrict__ hist, unsigned* __restrict__ ctrl,
                        const int* __restrict__ kptr) {
    __shared__ unsigned csum[256];
    const int tid = threadIdx.x;
    unsigned s = 0;
    for (int b = 0; b < 256; ++b) s += hist[tid * 256 + b];
    csum[tid] = s;
    __syncthreads();
    if (tid == 0) {
        const unsigned k = (unsigned)*kptr;
        unsigned cum = ctrl[1];
        int c = 255;
        for (; c > 0; --c) { if (cum + csum[c] >= k) break; cum += csum[c]; }
        int b = c * 256 + 255;
        for (; b > c * 256; --b) { if (cum + hist[b] >= k) break; cum += hist[b]; }
        ctrl[2] = (ctrl[0] << 16) | (unsigned)b;  // exact threshold key
        ctrl[3] = k - cum;                        // # ties to accept
        ctrl[4] = 0u;                             // tie-claim counter
    }
}

__global__ void k_spikify(float* __restrict__ buf, unsigned* __restrict__ ctrl,
                          size_t n) {
    const unsigned tk  = ctrl[2];
    const unsigned rem = ctrl[3];
    size_t stride = (size_t)gridDim.x * blockDim.x;
    for (size_t i = (size_t)blockIdx.x * blockDim.x + threadIdx.x; i < n; i += stride) {
        unsigned key = monokey(buf[i]);
        float sp = 0.0f;
        if (key > tk) sp = 1.0f;
        else if (key == tk) { if (atomicAdd(&ctrl[4], 1u) < rem) sp = 1.0f; }
        buf[i] = sp;   // logits region becomes the spikes output, in place
    }
}

// ---------------------------------------------------------------------------
// Backward: est[i,t] = b_tra[i] + sum_{o,d} spikes[o, t+15-d] * W[o,i,d]
// K = 1920 in 4 o-chunks of 32 (480 each); same WMMA micro-structure.
// ---------------------------------------------------------------------------
__global__ __launch_bounds__(256) void bwd_conv_wmma(
    const float* __restrict__ spikes, const float* __restrict__ W,
    const float* __restrict__ btra, float* __restrict__ est_out, int T)
{
    __shared__ float sWT[16 * WSTR];   // 16 i-rows x 480 of current o-chunk
    __shared__ float sS[WIN * TSTR];   // transposed spike window: [x][ol]

    const int t0  = blockIdx.x * COLS;
    const int i0  = blockIdx.y * 16;
    const int tid = threadIdx.x;
    const int lane = tid & 31;
    const int w    = tid >> 5;
    const int half = lane >> 4;
    const int ln   = lane & 15;
    const int col  = w * 16 + ln;

    const float* aBase = sWT + ln * WSTR + (half << 1);
    const float* bBase = sS + (col + 14) * TSTR + (half << 1);  // x = col+14-d

    v8f c = {};
    for (int oc = 0; oc < N_NEU; oc += 32) {
        __syncthreads();
        for (int idx = tid; idx < 16 * KDIM; idx += 256) {
            int m = idx / KDIM, kl = idx - m * KDIM;
            int ol = kl & 31, d = kl >> 5;            // kl = d*32 + ol
            sWT[m * WSTR + kl] = W[(size_t)(oc + ol) * KDIM + (i0 + m) * NDEL + d];
        }
        for (int idx = tid; idx < 32 * WIN; idx += 256) {
            int ol = idx / WIN, x = idx - ol * WIN;   // consecutive tid -> consecutive x
            int g = t0 + 1 + x;                       // window [t0+1, t0+142]
            sS[x * TSTR + ol] = (g < T) ? spikes[(size_t)(oc + ol) * T + g] : 0.0f;
        }
        __syncthreads();

        #pragma unroll
        for (int d = 0; d < NDEL; ++d) {
            #pragma unroll
            for (int ii = 0; ii < 32; ii += 4) {
                const v2f av = *(const v2f*)(aBase + (d << 5) + ii);
                const v2f bv = *(const v2f*)(bBase - d * TSTR + ii);
                c = __builtin_amdgcn_wmma_f32_16x16x4_f32(false, av, false, bv,
                                                          (short)0, c, false, false);
            }
        }
    }

    const int t = t0 + col;
    if (t < T) {
        #pragma unroll
        for (int r = 0; r < 8; ++r) {
            const int i = i0 + r + (half << 3);
            est_out[(size_t)i * T + t] = c[r] + btra[i];
        }
    }
}

// jnp.roll wrap: est[:, T-1] = b_tra + sum_o spikes[o,0] * W[o,:,0]
__global__ void k_fix_last(const float* __restrict__ spikes, const float* __restrict__ W,
                           const float* __restrict__ btra, float* __restrict__ est, int T)
{
    int i = threadIdx.x;
    if (i < N_PRE) {
        float acc = btra[i];
        for (int o = 0; o < N_NEU; ++o)
            acc += spikes[(size_t)o * T] * W[(size_t)o * KDIM + i * NDEL];
        est[(size_t)i * T + (T - 1)] = acc;
    }
}

// ---------------------------------------------------------------------------
extern "C" void kernel_launch(void* const* d_in, const int* in_sizes, int n_in,
                              void* d_out, int out_size, void* d_ws, size_t ws_size,
                              hipStream_t stream)
{
    const float* a     = (const float*)d_in[0];
    const float* W     = (const float*)d_in[1];
    const float* bconv = (const float*)d_in[2];
    const float* btra  = (const float*)d_in[3];
    const int*   kptr  = (const int*)d_in[4];     // k lives on device

    const int T = in_sizes[0] / N_PRE;
    const size_t n = (size_t)N_NEU * T;

    float* p_out  = (float*)d_out;                // p_b   : 128*T
    float* spikes = p_out + n;                    // spikes: 128*T (logits first)
    float* est    = spikes + n;                   // est   : 32*T

    unsigned* hist1 = (unsigned*)d_ws;            // 65536
    unsigned* hist2 = hist1 + 65536;              // 65536
    unsigned* ctrl  = hist2 + 65536;              // 8 control words

    const int colBlocks = (T + COLS - 1) / COLS;
    const int selGrid = 4096;

    // 1) forward conv (WMMA f32) -> p_b + stashed logits
    fwd_conv_wmma<<<dim3(colBlocks, N_NEU / 16), 256, 0, stream>>>(
        a, W, bconv, p_out, spikes, T);

    // 2) exact top-k radix select on logits
    k_zero<<<(131072 + 8 + 255) / 256, 256, 0, stream>>>(hist1, 131072 + 8);
    k_hist_hi<<<selGrid, 256, 0, stream>>>(spikes, hist1, n);
    k_scan1<<<1, 256, 0, stream>>>(hist1, ctrl, kptr);
    k_hist_lo<<<selGrid, 256, 0, stream>>>(spikes, hist2, ctrl, n);
    k_scan2<<<1, 256, 0, stream>>>(hist2, ctrl, kptr);
    k_spikify<<<selGrid, 256, 0, stream>>>(spikes, ctrl, n);

    // 3) backward conv (WMMA f32) -> est, then roll-wrap fixup of column T-1
    bwd_conv_wmma<<<dim3(colBlocks, N_PRE / 16), 256, 0, stream>>>(
        spikes, W, btra, est, T);
    k_fix_last<<<1, 32, 0, stream>>>(spikes, W, btra, est, T);
}